// GraphAttention_4217657885155
// MI455X (gfx1250) — compile-verified
//
#include <hip/hip_runtime.h>

#define N_NODES 100000
#define N_EDGES 1600000
#define F_IN 128
#define UNITS 32

typedef float v2f __attribute__((ext_vector_type(2)));
typedef float v8f __attribute__((ext_vector_type(8)));

// ---------------------------------------------------------------------------
// Kernel 1: H = node_states @ kernel  via V_WMMA_F32_16X16X4_F32 (fp32 WMMA)
// One wave32 per 16-row tile; B (128x32) staged in LDS, swizzled so that the
// two K-values a lane needs for a given column are adjacent (one ds_load_b64).
// A fragment (16x4 f32): lane<16 -> (row=lane, K=k0..k0+1),
//                        lane>=16 -> (row=lane-16, K=k0+2..k0+3)  => float2 load.
// ---------------------------------------------------------------------------
__global__ __launch_bounds__(256) void gat_gemm(const float* __restrict__ A,
                                                const float* __restrict__ K,
                                                float* __restrict__ H) {
  __shared__ float ldsB[F_IN * UNITS]; // 4096 floats, swizzled: [(k>>1)*64 + c*2 + (k&1)]
  for (int i = threadIdx.x; i < F_IN * UNITS; i += 256) {
    int k = i >> 5, c = i & 31;
    ldsB[(k >> 1) * 64 + c * 2 + (k & 1)] = K[i];
  }
  __syncthreads();

  const int wave = threadIdx.x >> 5;
  const int lane = threadIdx.x & 31;
  const int lo = lane & 15;
  const int hi = lane >> 4;

  const int numTiles = N_NODES / 16; // 6250, exact
  const int t = blockIdx.x * 8 + wave;
  if (t >= numTiles) return; // wave-uniform: EXEC stays all-ones for WMMA

  const float* arow = A + (size_t)(t * 16 + lo) * F_IN + 2 * hi;
  v8f c0 = {}; // output cols 0..15
  v8f c1 = {}; // output cols 16..31

#pragma unroll 4
  for (int k0 = 0; k0 < F_IN; k0 += 4) {
    v2f a = *(const v2f*)(arow + k0);
    const float* brow = ldsB + (k0 / 2 + hi) * 64;
    v2f b0 = *(const v2f*)(brow + lo * 2);
    v2f b1 = *(const v2f*)(brow + (lo + 16) * 2);
    c0 = __builtin_amdgcn_wmma_f32_16x16x4_f32(false, a, false, b0, (short)0, c0,
                                               false, false);
    c1 = __builtin_amdgcn_wmma_f32_16x16x4_f32(false, a, false, b1, (short)0, c1,
                                               false, false);
  }

  // C/D layout: VGPR r, lanes 0-15 -> (M=r, N=lane); lanes 16-31 -> (M=8+r, N=lane-16)
  float* hout = H + (size_t)(t * 16 + 8 * hi) * UNITS + lo;
#pragma unroll
  for (int r = 0; r < 8; ++r) {
    hout[r * UNITS] = c0[r];
    hout[r * UNITS + 16] = c1[r];
  }
}

// ---------------------------------------------------------------------------
// Kernel 2: per-node attention projections
//   p_src[n] = dot(h[n], wa[0:32]) ; p_dst[n] = dot(h[n], wa[32:64])
// ---------------------------------------------------------------------------
__global__ __launch_bounds__(256) void gat_proj(const float* __restrict__ H,
                                                const float* __restrict__ WA,
                                                float* __restrict__ P) {
  int n = blockIdx.x * blockDim.x + threadIdx.x;
  if (n >= N_NODES) return;
  const float* hr = H + (size_t)n * UNITS;
  float ps = 0.f, pd = 0.f;
#pragma unroll
  for (int u = 0; u < UNITS; ++u) {
    float hv = hr[u];
    ps = fmaf(hv, WA[u], ps);        // scalar-broadcast weight loads
    pd = fmaf(hv, WA[UNITS + u], pd);
  }
  P[n] = ps;
  P[N_NODES + n] = pd;
}

// ---------------------------------------------------------------------------
// Kernel 3: per-edge raw scores:  exp(clip(leaky_relu(p_src[s]+p_dst[d])))
// ---------------------------------------------------------------------------
__global__ __launch_bounds__(256) void gat_score(const float* __restrict__ P,
                                                 const int* __restrict__ esrc,
                                                 const int* __restrict__ edst,
                                                 float* __restrict__ S) {
  int e = blockIdx.x * blockDim.x + threadIdx.x;
  if (e >= N_EDGES) return;
  float x = P[esrc[e]] + P[N_NODES + edst[e]];
  x = (x > 0.f) ? x : 0.2f * x;           // leaky_relu slope 0.2
  x = fminf(2.f, fmaxf(-2.f, x));         // clip
  S[e] = expf(x);
}

__device__ __forceinline__ int lower_bound(const int* __restrict__ a, int n,
                                           int key) {
  int lo = 0, hi = n;
  while (lo < hi) {
    int mid = (lo + hi) >> 1;
    if (a[mid] < key) lo = mid + 1; else hi = mid;
  }
  return lo;
}

// ---------------------------------------------------------------------------
// Kernel 4: segment softmax + weighted aggregation. edge_src is SORTED, so one
// wave32 per node binary-searches its edge run; lane = output feature.
// No atomics: denominator via wave32 butterfly reduce.
// ---------------------------------------------------------------------------
__global__ __launch_bounds__(256) void gat_out(const float* __restrict__ H,
                                               const float* __restrict__ S,
                                               const int* __restrict__ esrc,
                                               const int* __restrict__ edst,
                                               float* __restrict__ out) {
  int node = blockIdx.x * 8 + (threadIdx.x >> 5);
  if (node >= N_NODES) return;
  int lane = threadIdx.x & 31;

  int e0 = lower_bound(esrc, N_EDGES, node);
  int e1 = lower_bound(esrc, N_EDGES, node + 1);

  // softmax denominator over the segment
  float denom = 0.f;
  for (int e = e0 + lane; e < e1; e += 32) denom += S[e];
#pragma unroll
  for (int off = 16; off > 0; off >>= 1) denom += __shfl_xor(denom, off, 32);

  // weighted feature aggregation: lane owns feature `lane`;
  // the 32 lanes together read one coalesced 128B row of H per edge (L2-hot).
  float acc = 0.f;
  for (int e = e0; e < e1; ++e) {
    acc = fmaf(S[e], H[(size_t)edst[e] * UNITS + lane], acc);
  }
  out[(size_t)node * UNITS + lane] = (e1 > e0) ? (acc / denom) : 0.f;
}

// ---------------------------------------------------------------------------
extern "C" void kernel_launch(void* const* d_in, const int* in_sizes, int n_in,
                              void* d_out, int out_size, void* d_ws,
                              size_t ws_size, hipStream_t stream) {
  const float* node_states = (const float*)d_in[0]; // [N,128]
  const float* kmat        = (const float*)d_in[1]; // [128,32]
  const float* katt        = (const float*)d_in[2]; // [64,1]
  const int*   esrc        = (const int*)d_in[3];   // [E] sorted
  const int*   edst        = (const int*)d_in[4];   // [E]
  float* out = (float*)d_out;                       // [N,32]

  // workspace layout: H [N*32] | P [2N] | S [E]  ~= 20 MB
  float* H = (float*)d_ws;
  float* P = H + (size_t)N_NODES * UNITS;
  float* S = P + 2 * (size_t)N_NODES;

  gat_gemm <<<(N_NODES / 16 + 7) / 8, 256, 0, stream>>>(node_states, kmat, H);
  gat_proj <<<(N_NODES + 255) / 256, 256, 0, stream>>>(H, katt, P);
  gat_score<<<(N_EDGES + 255) / 256, 256, 0, stream>>>(P, esrc, edst, S);
  gat_out  <<<(N_NODES + 7) / 8, 256, 0, stream>>>(H, S, esrc, edst, out);
}